// Encoder_39015482916864
// MI455X (gfx1250) — compile-verified
//
#include <hip/hip_runtime.h>
#include <hip/hip_bf16.h>

typedef float v2f __attribute__((ext_vector_type(2)));
typedef float v8f __attribute__((ext_vector_type(8)));

#define T_LEN    512
#define B_TOT    64
#define E_DIM    256
#define U_DIM    512
#define ZCOLS    2048              // 4*U
#define KTOT     768               // E + U
#define KSTEPS   192               // KTOT/4
#define A_STRIDE 772               // 768 + 4 pad -> conflict-free ds_load_b64
#define Z_STRIDE 260               // 256 + 4 pad
#define NCT      128               // 2048/16 column tiles per direction

// ---------------- ws layout ----------------
// [0, 256)                      : group barriers (8 groups * {count,gen}, 32B apart)
// [256, 256+131072*4)           : h ping-pong buffers: 8 groups * 2 * 16 * 512 floats
// [524544, +12582912)           : packed B fragments: 2 dirs * 128 ct * 192 ki * 32 lanes * float2
#define HBUF_OFF_BYTES   256
#define HBUF_FLOATS      (8 * 2 * 16 * 512)
#define BPACK_OFF_BYTES  (HBUF_OFF_BYTES + HBUF_FLOATS * 4)
#define BPACK_ELEMS      ((size_t)2 * NCT * KSTEPS * 32)   // float2 elements

__device__ __forceinline__ float sigm(float x) {
    return 1.0f / (1.0f + __expf(-x));
}

__device__ __forceinline__ void spin_pause() {
#if defined(__HIP_DEVICE_COMPILE__)
    __builtin_amdgcn_s_sleep(2);
#endif
}

// Zero the barrier + h-buffer region (64 + 131072 dwords).
__global__ void init_ws_kernel(unsigned* __restrict__ ws) {
    int i = blockIdx.x * 256 + threadIdx.x;
    if (i < 64 + HBUF_FLOATS) ws[i] = 0u;
}

// Pack [W;U] (768 x 2048, fp32) into per-lane WMMA B-fragment order:
// bpack[((d*128 + ct)*192 + ki)*32 + lane] = { B[4ki+2*(l>>4)][ct*16+(l&15)],
//                                             B[4ki+2*(l>>4)+1][ same col    ] }
__global__ void pack_weights_kernel(const float* __restrict__ Wf,
                                    const float* __restrict__ Uf,
                                    const float* __restrict__ Wb,
                                    const float* __restrict__ Ub,
                                    float2* __restrict__ bpack) {
    size_t idx = (size_t)blockIdx.x * 256 + threadIdx.x;
    if (idx >= BPACK_ELEMS) return;
    int l  = (int)(idx & 31);
    size_t rest = idx >> 5;
    int ki = (int)(rest % KSTEPS);
    rest /= KSTEPS;
    int ct = (int)(rest % NCT);
    int d  = (int)(rest / NCT);
    const float* W = d ? Wb : Wf;
    const float* U = d ? Ub : Uf;
    int c = ct * 16 + (l & 15);
    int k = ki * 4 + 2 * (l >> 4);
    float xv, yv;
    if (k < E_DIM) {
        xv = W[(size_t)k * ZCOLS + c];
        yv = W[(size_t)(k + 1) * ZCOLS + c];
    } else {
        xv = U[(size_t)(k - E_DIM) * ZCOLS + c];
        yv = U[(size_t)(k - E_DIM + 1) * ZCOLS + c];
    }
    bpack[idx] = make_float2(xv, yv);
}

__global__ __launch_bounds__(256, 1)
void lstm_persistent_kernel(const int*   __restrict__ x,
                            const float* __restrict__ emb,
                            const float* __restrict__ bias_f,
                            const float* __restrict__ bias_b,
                            const float* __restrict__ bpack_raw,
                            float*       __restrict__ hbuf,
                            unsigned*    __restrict__ bar,
                            float*       __restrict__ out) {
    __shared__ float lds_a[16 * A_STRIDE];   // A staging; reused as z (16 x Z_STRIDE)
    __shared__ float c_own[16 * 64];
    __shared__ float h_own[16 * 64];

    const int tid  = threadIdx.x;
    const int lane = tid & 31;
    const int w    = tid >> 5;

    const int bx    = blockIdx.x;
    const int ut    = bx & 7;          // unit tile (64 units)
    const int btile = (bx >> 3) & 3;   // batch tile (16 rows)
    const int dir   = bx >> 5;         // 0 = fwd, 1 = bwd
    const int grp   = dir * 4 + btile;
    const int b0    = btile * 16;
    const int j0    = ut * 64;

    // per-wave column assignment: wave w covers local z cols [w*32, w*32+32)
    const int gate  = w >> 1;
    const int lc0   = w * 32 + (lane & 15);
    const int gcol0 = gate * 512 + j0 + (w & 1) * 32 + (lane & 15);
    const float* bias = dir ? bias_b : bias_f;
    const float bias0 = bias[gcol0];
    const float bias1 = bias[gcol0 + 16];

    const int ct0 = (gate * 512 + j0 + (w & 1) * 32) >> 4;
    const v2f* bp0 = (const v2f*)bpack_raw + ((size_t)(dir * NCT + ct0) * KSTEPS) * 32 + lane;
    const v2f* bp1 = bp0 + (size_t)KSTEPS * 32;   // ct0+1

    const int arow = (lane & 15) * A_STRIDE + 2 * (lane >> 4);

    float* hg = hbuf + (size_t)grp * 2 * 16 * 512;
    unsigned* bcnt = bar + grp * 8;
    unsigned* bgen = bcnt + 4;
    unsigned gen_target = 0;

    // zero persistent c/h slices
    for (int i = tid; i < 16 * 64; i += 256) { c_own[i] = 0.0f; h_own[i] = 0.0f; }
    __syncthreads();

    for (int t = 0; t < T_LEN; ++t) {
        const int t_eff = dir ? (T_LEN - 1 - t) : t;
        const float* hcur = hg + (t & 1) * (16 * 512);
        float*       hnxt = hg + ((t + 1) & 1) * (16 * 512);

        // ---- stage A = [emb_row | h_prev] into LDS ----
        {
            const int r  = tid >> 4;
            const int c16 = tid & 15;
            const int tok = x[(b0 + r) * T_LEN + t_eff];
            const float4* esrc = (const float4*)(emb + (size_t)tok * E_DIM) + c16 * 4;
            float4* edst = (float4*)(lds_a + r * A_STRIDE) + c16 * 4;
#pragma unroll
            for (int i = 0; i < 4; ++i) edst[i] = esrc[i];
            const float4* hsrc = (const float4*)(hcur + r * U_DIM) + c16 * 8;
            float4* hdst = (float4*)(lds_a + r * A_STRIDE + E_DIM) + c16 * 8;
#pragma unroll
            for (int i = 0; i < 8; ++i) hdst[i] = hsrc[i];
        }
        __syncthreads();

        // ---- WMMA GEMM: two 16x16 output tiles per wave, K = 768 ----
        v8f acc0 = {bias0, bias0, bias0, bias0, bias0, bias0, bias0, bias0};
        v8f acc1 = {bias1, bias1, bias1, bias1, bias1, bias1, bias1, bias1};
#if defined(__HIP_DEVICE_COMPILE__)
#pragma unroll 4
        for (int ki = 0; ki < KSTEPS; ++ki) {
            v2f a  = *(const v2f*)(lds_a + arow + ki * 4);
            v2f bA = bp0[(size_t)ki * 32];
            v2f bB = bp1[(size_t)ki * 32];
            acc0 = __builtin_amdgcn_wmma_f32_16x16x4_f32(false, a, false, bA,
                                                         (short)0, acc0, false, false);
            acc1 = __builtin_amdgcn_wmma_f32_16x16x4_f32(false, a, false, bB,
                                                         (short)0, acc1, false, false);
        }
#endif
        __syncthreads();   // everyone done reading A before z overwrites it

        // ---- spill z tiles to LDS (C/D layout: vgpr v -> row v + 8*(lane>>4)) ----
        {
            const int rb = 8 * (lane >> 4);
#pragma unroll
            for (int v = 0; v < 8; ++v) {
                lds_a[(v + rb) * Z_STRIDE + lc0]      = acc0[v];
                lds_a[(v + rb) * Z_STRIDE + lc0 + 16] = acc1[v];
            }
        }
        __syncthreads();

        // ---- gate math + state update: thread -> (row, 4 units) ----
        {
            const int r = tid >> 4;
            const int j = (tid & 15) * 4;
            const bool m = x[(b0 + r) * T_LEN + t_eff] != 0;
            const int zb = r * Z_STRIDE + j;
            const int sb = r * 64 + j;
            float hv[4], cv[4];
#pragma unroll
            for (int q = 0; q < 4; ++q) {
                const float vi = sigm(lds_a[zb + q]);
                const float vf = sigm(lds_a[zb + 64 + q]);
                const float vg = tanhf(lds_a[zb + 128 + q]);
                const float vo = sigm(lds_a[zb + 192 + q]);
                const float co = c_own[sb + q];
                const float ho = h_own[sb + q];
                const float cn = vf * co + vi * vg;
                const float hn = vo * tanhf(cn);
                const float cs = m ? cn : co;
                const float hs = m ? hn : ho;
                c_own[sb + q] = cs;
                h_own[sb + q] = hs;
                cv[q] = cs; hv[q] = hs;
            }
            const float4 h4 = make_float4(hv[0], hv[1], hv[2], hv[3]);
            *(float4*)(hnxt + r * U_DIM + j0 + j) = h4;
            // encoder_output[b][t_eff][dir*512 + j0 + j]
            *(float4*)(out + ((size_t)(b0 + r) * T_LEN + t_eff) * 1024 + dir * 512 + j0 + j) = h4;
            if (t == T_LEN - 1) {
                const size_t enc = (size_t)B_TOT * T_LEN * 1024;
                *(float4*)(out + enc + (size_t)(b0 + r) * 1024 + dir * 512 + j0 + j) = h4;
                *(float4*)(out + enc + (size_t)B_TOT * 1024 +
                           (size_t)(b0 + r) * 1024 + dir * 512 + j0 + j) =
                    make_float4(cv[0], cv[1], cv[2], cv[3]);
            }
        }

        // ---- group barrier across the 8 unit-tile WGs of (dir, btile) ----
        __threadfence();
        __syncthreads();
        ++gen_target;
        if (tid == 0) {
            unsigned old = __hip_atomic_fetch_add(bcnt, 1u, __ATOMIC_ACQ_REL,
                                                  __HIP_MEMORY_SCOPE_AGENT);
            if (old == 7u) {
                __hip_atomic_store(bcnt, 0u, __ATOMIC_RELAXED, __HIP_MEMORY_SCOPE_AGENT);
                __hip_atomic_fetch_add(bgen, 1u, __ATOMIC_RELEASE, __HIP_MEMORY_SCOPE_AGENT);
            }
            while (__hip_atomic_load(bgen, __ATOMIC_ACQUIRE, __HIP_MEMORY_SCOPE_AGENT) < gen_target)
                spin_pause();
        }
        __syncthreads();
    }
}

extern "C" void kernel_launch(void* const* d_in, const int* in_sizes, int n_in,
                              void* d_out, int out_size, void* d_ws, size_t ws_size,
                              hipStream_t stream) {
    const int*   x   = (const int*)d_in[0];
    const float* emb = (const float*)d_in[1];
    const float* Wf  = (const float*)d_in[2];
    const float* Uf  = (const float*)d_in[3];
    const float* bf  = (const float*)d_in[4];
    const float* Wb  = (const float*)d_in[5];
    const float* Ub  = (const float*)d_in[6];
    const float* bb  = (const float*)d_in[7];
    float* out = (float*)d_out;

    float* hbuf  = (float*)((char*)d_ws + HBUF_OFF_BYTES);
    float* bpack = (float*)((char*)d_ws + BPACK_OFF_BYTES);
    (void)in_sizes; (void)n_in; (void)out_size; (void)ws_size;

    // 1) zero barriers + h ping-pong buffers (fresh every call: graph-safe)
    init_ws_kernel<<<(64 + HBUF_FLOATS + 255) / 256, 256, 0, stream>>>((unsigned*)d_ws);
    // 2) pack weights into per-lane WMMA fragment order
    pack_weights_kernel<<<(int)((BPACK_ELEMS + 255) / 256), 256, 0, stream>>>(
        Wf, Uf, Wb, Ub, (float2*)bpack);
    // 3) persistent bidirectional LSTM: 2 dirs * 4 batch-tiles * 8 unit-tiles
    lstm_persistent_kernel<<<64, 256, 0, stream>>>(x, emb, bf, bb, bpack,
                                                   hbuf, (unsigned*)d_ws, out);
}